// LlamaDecodeAttention_82068235091993
// MI455X (gfx1250) — compile-verified
//
#include <hip/hip_runtime.h>
#include <hip/hip_bf16.h>
#include <math.h>

// ---------------------------------------------------------------------------
// Llama decode attention for MI455X (gfx1250, wave32, WMMA).
//
// Roofline: ~440MB moved per call (W_qkv 100MB + W_o 67MB + K/V gather ~270MB)
// => ~19us at 23.3TB/s HBM; FLOPs (~3.8G fp32) are negligible. So:
//  * fp32 end-to-end with native V_WMMA_F32_16X16X4_F32 (matches reference
//    precision; matrix ops are "free" at this intensity).
//  * Split-K x8 GEMMs for enough in-flight loads to saturate HBM
//    (3072 waves for QKV instead of 384); deterministic two-pass reduce.
//  * K/V streams use non-temporal loads so the 167MB of weights can stay
//    resident in the 192MB L2 across graph replays.
// ---------------------------------------------------------------------------

#define H_      32
#define KVH_    8
#define D_      128
#define HIDDEN_ 4096
#define QS_     4096
#define KVS_    1024
#define QKV_    6144
#define MAXBPS_ 128
#define BATCH_  32
#define KSPLIT_ 8
#define KCHUNK_ (HIDDEN_ / KSPLIT_)   // 512
#define NEG_BIG (-3.0e38f)

typedef __attribute__((ext_vector_type(2))) float v2f;
typedef __attribute__((ext_vector_type(8))) float v8f;

__device__ __forceinline__ v8f wmma_f32(v2f a, v2f b, v8f c) {
  // D(16x16,f32) = A(16x4,f32) * B(4x16,f32) + C
  return __builtin_amdgcn_wmma_f32_16x16x4_f32(
      /*neg_a=*/false, a, /*neg_b=*/false, b,
      /*c_mod=*/(short)0, c, /*reuse_a=*/false, /*reuse_b=*/false);
}

// ---------------------------------------------------------------------------
// Split-K GEMM: Cpart[ks][32 x N] = A[32 x k-chunk] * W[N x k-chunk]^T
// One wave computes a 32x16 tile over a 512-wide K chunk (two accumulators
// share each B load, so W is still streamed exactly once in total).
// grid = (N/64, KSPLIT), block = 128 (4 waves).
// ---------------------------------------------------------------------------
__global__ __launch_bounds__(128) void gemm32_nt_wmma_ksplit(
    const float* __restrict__ A, const float* __restrict__ W,
    float* __restrict__ Cpart, int N)
{
  const int lane = threadIdx.x & 31;
  const int wave = threadIdx.x >> 5;
  const int half = lane >> 4;
  const int l16  = lane & 15;
  const int n0   = (blockIdx.x * 4 + wave) * 16;
  const int k0   = blockIdx.y * KCHUNK_;

  // A layout (16x4 f32): lane L -> row M = L%16; VGPR0/1 -> K = 2*(L/16)+{0,1}
  const float* a0 = A + (size_t)l16 * HIDDEN_ + k0;          // rows 0..15
  const float* a1 = A + (size_t)(l16 + 16) * HIDDEN_ + k0;   // rows 16..31
  // B layout (4x16 f32): lane L -> col N = L%16; K = 2*(L/16)+{0,1}
  const float* wr = W + (size_t)(n0 + l16) * HIDDEN_ + k0;

  v8f c0 = {0,0,0,0,0,0,0,0};
  v8f c1 = {0,0,0,0,0,0,0,0};

  #pragma unroll 4
  for (int k = 0; k < KCHUNK_; k += 4) {
    const int d = k + half * 2;
    v2f av0 = *(const v2f*)(a0 + d);
    v2f av1 = *(const v2f*)(a1 + d);
    v2f wv  = *(const v2f*)(wr + d);
    c0 = wmma_f32(av0, wv, c0);
    c1 = wmma_f32(av1, wv, c1);
  }

  // C layout: lane L -> col n0+L%16, rows M = r + 8*(L/16)
  float* outp = Cpart + (size_t)blockIdx.y * 32 * N;
  #pragma unroll
  for (int r = 0; r < 8; ++r) {
    const int m = r + 8 * half;
    outp[(size_t)m * N + n0 + l16]        = c0[r];
    outp[(size_t)(m + 16) * N + n0 + l16] = c1[r];
  }
}

// Cout[32 x N] = bias[N] + sum over KSPLIT partials (fixed order: determinate).
__global__ __launch_bounds__(256) void ksplit_reduce(
    const float* __restrict__ Cpart, const float* __restrict__ bias,
    float* __restrict__ Cout, int N)
{
  const int i = blockIdx.x * 256 + threadIdx.x;
  if (i >= 32 * N) return;
  const int n = i % N;
  float s = bias[n];
  #pragma unroll
  for (int p = 0; p < KSPLIT_; ++p) s += Cpart[(size_t)p * 32 * N + i];
  Cout[i] = s;
}

// ---------------------------------------------------------------------------
// RoPE in-place on ws_qkv (bias already added by the reduce pass).
// 40 heads (32 q + 8 k) x 64 rotation pairs per batch row.
// ---------------------------------------------------------------------------
__global__ __launch_bounds__(256) void rope_kernel(
    float* __restrict__ qkv, const int* __restrict__ ctx)
{
  const int t = blockIdx.x * blockDim.x + threadIdx.x;
  if (t >= BATCH_ * 40 * 64) return;
  const int b  = t / (40 * 64);
  const int r  = t - b * (40 * 64);
  const int hd = r >> 6;
  const int d  = r & 63;
  float* base = qkv + (size_t)b * QKV_ +
                (hd < H_ ? hd * D_ : QS_ + (hd - H_) * D_);
  const float pos = (float)(ctx[b] - 1);
  const float inv = powf(10000.0f, -(float)d * (1.0f / 64.0f));
  const float ang = pos * inv;
  const float cs = cosf(ang), sn = sinf(ang);
  const float x1 = base[d], x2 = base[d + 64];
  base[d]      = x1 * cs - x2 * sn;
  base[d + 64] = x2 * cs + x1 * sn;
}

// ---------------------------------------------------------------------------
// Paged GQA decode attention. One block per (kv-head, batch); 8 waves split
// the ctx tiles (16 tokens == 1 cache block) with online softmax, flash-style
// cross-wave reduction in LDS. Both scores and P*V use f32 WMMA.
// K/V cache reads are non-temporal: pure streaming traffic that must not
// evict the L2-resident weight matrices.
// Score tile:  C(16 tok x 16) = K_tile(16x128) * Q^T(128x16)  [heads 0..3 live]
// PV tile:     C(16 hd  x 16) = P(16x16tok)    * V(16tok x 16d) per 16-d slab
// ---------------------------------------------------------------------------
__global__ __launch_bounds__(256) void attn_decode_wmma(
    const float* __restrict__ qkv, const float* __restrict__ kc,
    const float* __restrict__ vc,  const int* __restrict__ bt,
    const int* __restrict__ ctxl,  float* __restrict__ out)
{
  __shared__ float qs[16 * D_];          // q group, scaled; heads>=4 zeroed
  __shared__ float pl[8][256];           // per-wave P tile [token][head]
  __shared__ float red_m[8][4];
  __shared__ float red_l[8][4];
  __shared__ float red_acc[8][4][D_];

  const int kvh  = blockIdx.x;
  const int b    = blockIdx.y;
  const int tid  = threadIdx.x;
  const int lane = tid & 31;
  const int wave = tid >> 5;
  const int half = lane >> 4;
  const int l16  = lane & 15;

  const int ctx = ctxl[b];
  const int pos = ctx - 1;
  const float scaling = 0.088388347648318447f;  // 1/sqrt(128)

  const float* qbase = qkv + (size_t)b * QKV_;
  for (int i = tid; i < 16 * D_; i += 256)
    qs[i] = (i < 4 * D_) ? qbase[kvh * 4 * D_ + i] * scaling : 0.0f;
  __syncthreads();

  // fresh (post-RoPE) k/v for the newest token -- global ws, so the s==pos
  // select stays a global/global pointer select (no flat path).
  const float* kfresh = qkv + (size_t)b * QKV_ + QS_ + kvh * D_;
  const float* vfresh = qkv + (size_t)b * QKV_ + QS_ + KVS_ + kvh * D_;

  float mc = NEG_BIG;   // running max of this lane's head column
  float lc = 0.0f;      // running sum of this lane's head column
  v8f acc[8];
  #pragma unroll
  for (int i = 0; i < 8; ++i) acc[i] = (v8f){0,0,0,0,0,0,0,0};

  const int ntiles = (ctx + 15) >> 4;
  for (int t = wave; t < ntiles; t += 8) {
    const int blk = bt[b * MAXBPS_ + t];
    const int tok = l16;                 // A-operand token row
    const int sA  = t * 16 + tok;
    const float* kp = (sA == pos)
        ? kfresh
        : kc + ((size_t)(blk * 16 + tok) * KVH_ + kvh) * D_;

    // ---- scores: 32 k-steps over D=128 (NT: streaming) ----
    v8f sc = (v8f){0,0,0,0,0,0,0,0};
    #pragma unroll
    for (int ks = 0; ks < 32; ++ks) {
      const int d = ks * 4 + half * 2;
      v2f av = __builtin_nontemporal_load((const v2f*)(kp + d));
      v2f qv;
      qv.x = qs[l16 * D_ + d];
      qv.y = qs[l16 * D_ + d + 1];
      sc = wmma_f32(av, qv, sc);
    }

    // ---- mask + per-head (column) tile max ----
    float sr[8];
    float vmax = NEG_BIG;
    #pragma unroll
    for (int r = 0; r < 8; ++r) {
      const int s = t * 16 + r + 8 * half;   // token of C row
      const float x = (s < ctx) ? sc[r] : NEG_BIG;
      sr[r] = x;
      vmax = fmaxf(vmax, x);
    }
    vmax = fmaxf(vmax, __shfl_xor(vmax, 16, 32));

    const float nmc = fmaxf(mc, vmax);
    const float fac = __expf(mc - nmc);    // rescale for this lane's head
    mc = nmc;

    // ---- P = exp(S - m) -> LDS tile; accumulate row sums ----
    float rs = 0.0f;
    #pragma unroll
    for (int r = 0; r < 8; ++r) {
      const float p = __expf(sr[r] - mc);
      rs += p;
      pl[wave][(r + 8 * half) * 16 + l16] = p;  // [token][head]
    }
    rs += __shfl_xor(rs, 16, 32);
    lc = lc * fac + rs;

    // ---- rescale accumulators (row r holds head r + 8*half) ----
    float fr[8];
    #pragma unroll
    for (int r = 0; r < 8; ++r)
      fr[r] = __shfl(fac, r + 8 * half, 32);
    #pragma unroll
    for (int nt = 0; nt < 8; ++nt) {
      #pragma unroll
      for (int r = 0; r < 8; ++r) acc[nt][r] *= fr[r];
    }

    // ---- P * V : 4 token-chunks x 8 d-slabs (NT V stream) ----
    #pragma unroll
    for (int j = 0; j < 4; ++j) {
      const int t0 = 4 * j + 2 * half;
      const int t1 = t0 + 1;
      v2f pa;
      pa.x = pl[wave][t0 * 16 + l16];
      pa.y = pl[wave][t1 * 16 + l16];
      const int s0 = t * 16 + t0, s1 = t * 16 + t1;
      const float* v0 = (s0 == pos)
          ? vfresh : vc + ((size_t)(blk * 16 + t0) * KVH_ + kvh) * D_;
      const float* v1 = (s1 == pos)
          ? vfresh : vc + ((size_t)(blk * 16 + t1) * KVH_ + kvh) * D_;
      #pragma unroll
      for (int nt = 0; nt < 8; ++nt) {
        const int d = nt * 16 + l16;
        v2f bv;
        bv.x = __builtin_nontemporal_load(v0 + d);
        bv.y = __builtin_nontemporal_load(v1 + d);
        acc[nt] = wmma_f32(pa, bv, acc[nt]);
      }
    }
  }

  // ---- cross-wave flash reduction ----
  if (half == 0) {
    if (l16 < 4) { red_m[wave][l16] = mc; red_l[wave][l16] = lc; }
    #pragma unroll
    for (int nt = 0; nt < 8; ++nt) {
      #pragma unroll
      for (int r = 0; r < 4; ++r)
        red_acc[wave][r][nt * 16 + l16] = acc[nt][r];
    }
  }
  __syncthreads();

  for (int e = tid; e < 4 * D_; e += 256) {
    const int h2 = e >> 7;
    const int d  = e & (D_ - 1);
    float M = NEG_BIG;
    #pragma unroll
    for (int w = 0; w < 8; ++w) M = fmaxf(M, red_m[w][h2]);
    float so = 0.0f, sl = 0.0f;
    #pragma unroll
    for (int w = 0; w < 8; ++w) {
      const float f = __expf(red_m[w][h2] - M);
      so += f * red_acc[w][h2][d];
      sl += f * red_l[w][h2];
    }
    out[(size_t)b * HIDDEN_ + (kvh * 4 + h2) * D_ + d] = so / sl;
  }
}

// ---------------------------------------------------------------------------
extern "C" void kernel_launch(void* const* d_in, const int* in_sizes, int n_in,
                              void* d_out, int out_size, void* d_ws, size_t ws_size,
                              hipStream_t stream) {
  (void)in_sizes; (void)n_in; (void)out_size; (void)ws_size;
  const float* hidden = (const float*)d_in[0];
  const float* Wqkv   = (const float*)d_in[1];
  const float* bqkv   = (const float*)d_in[2];
  const float* Wo     = (const float*)d_in[3];
  const float* bo     = (const float*)d_in[4];
  const float* kc     = (const float*)d_in[5];
  const float* vc     = (const float*)d_in[6];
  const int*   bt     = (const int*)d_in[7];
  const int*   ctx    = (const int*)d_in[8];
  float* outp = (float*)d_out;

  float* ws_qkv  = (float*)d_ws;                         // 32 x 6144
  float* ws_attn = ws_qkv  + (size_t)BATCH_ * QKV_;      // 32 x 4096
  float* ws_part = ws_attn + (size_t)BATCH_ * HIDDEN_;   // 8 x 32 x 6144 max

  // 1) qkv = hidden @ W_qkv^T + b_qkv   (split-K x8, then reduce)
  gemm32_nt_wmma_ksplit<<<dim3(QKV_ / 64, KSPLIT_), dim3(128), 0, stream>>>(
      hidden, Wqkv, ws_part, QKV_);
  ksplit_reduce<<<dim3((32 * QKV_) / 256), dim3(256), 0, stream>>>(
      ws_part, bqkv, ws_qkv, QKV_);
  // 2) RoPE on q and k in-place
  rope_kernel<<<dim3((BATCH_ * 40 * 64 + 255) / 256), dim3(256), 0, stream>>>(
      ws_qkv, ctx);
  // 3) paged GQA decode attention -> ws_attn
  attn_decode_wmma<<<dim3(KVH_, BATCH_), dim3(256), 0, stream>>>(
      ws_qkv, kc, vc, bt, ctx, ws_attn);
  // 4) out = attn @ W_o^T + b_o   (split-K x8, then reduce)
  gemm32_nt_wmma_ksplit<<<dim3(HIDDEN_ / 64, KSPLIT_), dim3(128), 0, stream>>>(
      ws_attn, Wo, ws_part, HIDDEN_);
  ksplit_reduce<<<dim3((32 * HIDDEN_) / 256), dim3(256), 0, stream>>>(
      ws_part, bo, outp, HIDDEN_);
}